// AdjustableModePooling_3461743640934
// MI455X (gfx1250) — compile-verified
//
#include <hip/hip_runtime.h>

// AdjustableModePooling: mode (ties -> smallest) over consecutive groups of 8
// labels (0..9 stored as f32), with a b-major transpose on output.
//   input  flat: 64*65536*10 = 41,943,040 f32, group g = b*81920 + c*8192 + j
//   output flat: (c*8192 + j)*64 + b,  5,242,880 f32
// Memory-bound: ~189 MB traffic -> ~8us at 23.3 TB/s. No matrix math, so the
// CDNA5 feature used is the async global->LDS DMA path (ASYNCcnt) to decouple
// the read layout from the transposed write layout.

#define B64_  64
#define CDIM_ 10
#define NJ_   8192
#define OUT_PER_C_ (NJ_ * B64_)     // 524288 = 2^19
#define TPB_  256
#define CHUNK_STRIDE_ 144           // 128B data + 16B pad -> conflict-minimal ds_load_b128

__global__ __launch_bounds__(TPB_)
void AdjustableModePooling_kernel(const float* __restrict__ x,
                                  float* __restrict__ out,
                                  int total) {
    __shared__ __align__(16) char smem[B64_ * CHUNK_STRIDE_];  // 9216 B

    const int t      = threadIdx.x;
    const int o_base = blockIdx.x * TPB_;
    const int c      = o_base >> 19;                       // / (NJ*B)
    const int j0     = (o_base & (OUT_PER_C_ - 1)) >> 6;   // / B

    // ---- Stage 64 chunks x 128B of input into LDS via async DMA ----------
    // chunk m (== b): global bytes [(m*81920 + c*8192 + j0)*32, +128)
    const unsigned long long gbase = (unsigned long long)(const void*)x;
    const unsigned chunk_elem = (unsigned)c * 8192u + (unsigned)j0;
#pragma unroll
    for (int r = 0; r < 2; ++r) {
        const int m = r * 32 + (t >> 3);   // chunk index 0..63
        const int q = t & 7;               // 16B slice within 128B chunk
        unsigned long long gaddr = gbase
            + ((unsigned long long)((unsigned)m * 81920u + chunk_elem)) * 32ull
            + (unsigned)(q * 16);
        unsigned lds_addr = (unsigned)(size_t)(smem + m * CHUNK_STRIDE_ + q * 16);
        asm volatile("global_load_async_to_lds_b128 %0, %1, off"
                     :: "v"(lds_addr), "v"(gaddr)
                     : "memory");
    }
    asm volatile("s_wait_asynccnt 0" ::: "memory");
    __syncthreads();

    // ---- Compute: thread t handles output (b = t&63, j = j0 + t>>6) ------
    const int b = t & 63;
    const int j = t >> 6;
    const float4* p = (const float4*)(smem + b * CHUNK_STRIDE_ + j * 32);
    const float4 a0 = p[0];
    const float4 a1 = p[1];
    const float v[8] = {a0.x, a0.y, a0.z, a0.w, a1.x, a1.y, a1.z, a1.w};

    // Nibble histogram of labels 0..9 in one 64-bit register (counts <= 8,
    // so no nibble carry). Max shift = 9*4 = 36 < 64.
    unsigned long long acc = 0ull;
#pragma unroll
    for (int i = 0; i < 8; ++i) {
        unsigned iv = (unsigned)v[i];          // exact: labels are integral
        acc += 1ull << (iv * 4u);
    }

    // Smallest value with max count (torch.mode tie rule): strict '>' keeps
    // the first (smallest) label on ties.
    int      bestVal = 0;
    unsigned bestCnt = (unsigned)acc & 0xFu;
#pragma unroll
    for (int val = 1; val < 10; ++val) {
        unsigned cnt = (unsigned)(acc >> (4 * val)) & 0xFu;
        if (cnt > bestCnt) { bestCnt = cnt; bestVal = val; }
    }

    const int o = o_base + t;                  // fully coalesced store
    if (o < total) out[o] = (float)bestVal;
}

extern "C" void kernel_launch(void* const* d_in, const int* in_sizes, int n_in,
                              void* d_out, int out_size, void* d_ws, size_t ws_size,
                              hipStream_t stream) {
    (void)in_sizes; (void)n_in; (void)d_ws; (void)ws_size;
    const float* x = (const float*)d_in[0];
    float* out = (float*)d_out;
    const int blocks = (out_size + TPB_ - 1) / TPB_;   // 20480 for the given shapes
    AdjustableModePooling_kernel<<<blocks, TPB_, 0, stream>>>(x, out, out_size);
}